// GAT_23768349016464
// MI455X (gfx1250) — compile-verified
//
#include <hip/hip_runtime.h>
#include <math.h>

// ---------------------------------------------------------------------------
// 2-layer GAT forward for MI455X (gfx1250, wave32).
// GEMMs use V_WMMA_F32_16X16X4_F32 (fp32 in/out, matches reference precision);
// K and Nc are compile-time so the K-loop fully unrolls into a WMMA chain.
// Edge softmax/aggregation use float atomics (bandwidth-bound phase);
// H/C are template parameters so head indexing is shift/mask, not idiv.
// ---------------------------------------------------------------------------

typedef __attribute__((ext_vector_type(2))) float v2f;
typedef __attribute__((ext_vector_type(8))) float v8f;

#define GAT_NEG_SLOPE 0.2f
#define GAT_HC 128            // H*C is 128 for both layers

__device__ __forceinline__ float gat_leaky(float x) {
    return x > 0.f ? x : GAT_NEG_SLOPE * x;
}

__device__ __forceinline__ void gat_atomic_max_f32(float* addr, float val) {
    // IEEE-754 monotone bit trick: works for mixed signs with -inf init.
    if (val >= 0.f) {
        atomicMax((int*)addr, __float_as_int(val));
    } else {
        atomicMin((unsigned int*)addr, __float_as_uint(val));
    }
}

// ---------------------------------------------------------------------------
// fill: initialize workspace regions (graph-capture safe, no hipMemset)
// ---------------------------------------------------------------------------
__global__ void gat_fill_kernel(float* __restrict__ p, float v, int n) {
    int t = blockIdx.x * blockDim.x + threadIdx.x;
    if (t < n) p[t] = v;
}

// ---------------------------------------------------------------------------
// WMMA GEMM: C[M x 128] = A[M x K] * B[K x 128], fp32 row-major.
// One wave computes one 16x16 tile; K/4 chained 16x16x4 f32 WMMAs, fully
// unrolled. A-fragment loaded as a single b64; branchless row clamp keeps
// EXEC all-ones through the WMMA chain.
// ---------------------------------------------------------------------------
template <int K>
__global__ void gat_wmma_gemm_f32(const float* __restrict__ A,
                                  const float* __restrict__ B,
                                  float* __restrict__ C,
                                  int M) {
    constexpr int Nc = 128;
    const int lane    = threadIdx.x & 31;
    const int l16     = lane & 15;
    const int halfsel = lane >> 4;                 // 0 for lanes 0-15, 1 for 16-31
    const int wave    = blockIdx.x * (blockDim.x >> 5) + (threadIdx.x >> 5);

    const int tileM = wave >> 3;                   // Nc/16 == 8 column tiles
    const int tileN = wave & 7;
    if (tileM * 16 >= M) return;                   // uniform per wave

    const int row0 = tileM * 16;
    const int col0 = tileN * 16;
    int arow = row0 + l16;
    arow = arow < M ? arow : (M - 1);              // branchless clamp (M%16==0 here)

    const float* Aptr = A + (long long)arow * K + 2 * halfsel;
    const float* Bptr = B + (long long)(2 * halfsel) * Nc + col0 + l16;

    v8f acc = {};
    #pragma unroll
    for (int kk = 0; kk < K; kk += 4) {
        // A 16x4 frag: lane holds A[arow][kk+2h], A[arow][kk+2h+1] (contiguous)
        v2f a = *(const v2f*)(Aptr + kk);
        // B 4x16 frag: lane holds B[kk+2h][col], B[kk+2h+1][col]
        v2f b;
        b.x = Bptr[kk * Nc];
        b.y = Bptr[(kk + 1) * Nc];
        acc = __builtin_amdgcn_wmma_f32_16x16x4_f32(
            /*neg_a=*/false, a, /*neg_b=*/false, b,
            /*c_mod=*/(short)0, acc, /*reuse_a=*/false, /*reuse_b=*/false);
    }

    #pragma unroll
    for (int i = 0; i < 8; ++i) {
        const int r = row0 + i + 8 * halfsel;      // D layout: M = i + 8*halfsel
        if (r < M) C[(long long)r * Nc + col0 + l16] = acc[i];
    }
}

// ---------------------------------------------------------------------------
// Per-(node,head) attention logits: a_s[n,h] = <xp[n,h,:], att_src[h,:]>
// ---------------------------------------------------------------------------
template <int H>
__global__ void gat_att_kernel(const float* __restrict__ xp,
                               const float* __restrict__ att_s,
                               const float* __restrict__ att_d,
                               float* __restrict__ a_s,
                               float* __restrict__ a_d,
                               int Nn) {
    constexpr int C = GAT_HC / H;
    int t = blockIdx.x * blockDim.x + threadIdx.x;   // n*H + h
    if (t >= Nn * H) return;
    const int h = t & (H - 1);
    const int n = t / H;                             // H power-of-2 -> shift
    const float* xrow = xp + (long long)n * GAT_HC + h * C;
    const float* as = att_s + h * C;
    const float* ad = att_d + h * C;
    float ss = 0.f, sd = 0.f;
    #pragma unroll
    for (int c = 0; c < C; ++c) {
        float v = xrow[c];
        ss += v * as[c];
        sd += v * ad[c];
    }
    a_s[t] = ss;
    a_d[t] = sd;
}

// ---------------------------------------------------------------------------
// Edge pass 1: segment max of leaky_relu(a_src[src]+a_dst[dst]) over dst.
// One thread per (edge, head); self-loops appended after the E real edges.
// ---------------------------------------------------------------------------
template <int H>
__global__ void gat_edge_max_kernel(const int* __restrict__ ei, int E, int Nn,
                                    const float* __restrict__ a_s,
                                    const float* __restrict__ a_d,
                                    float* __restrict__ m) {
    int t = blockIdx.x * blockDim.x + threadIdx.x;
    const int total = (E + Nn) * H;
    if (t >= total) return;
    const int h = t & (H - 1);
    const int e = t / H;
    const int s = (e < E) ? ei[e]     : (e - E);
    const int d = (e < E) ? ei[E + e] : (e - E);
    const float alpha = gat_leaky(a_s[s * H + h] + a_d[d * H + h]);
    gat_atomic_max_f32(&m[d * H + h], alpha);
}

// ---------------------------------------------------------------------------
// Edge pass 2: denom[dst,h] += exp(alpha - m[dst,h])
// ---------------------------------------------------------------------------
template <int H>
__global__ void gat_edge_sum_kernel(const int* __restrict__ ei, int E, int Nn,
                                    const float* __restrict__ a_s,
                                    const float* __restrict__ a_d,
                                    const float* __restrict__ m,
                                    float* __restrict__ denom) {
    int t = blockIdx.x * blockDim.x + threadIdx.x;
    const int total = (E + Nn) * H;
    if (t >= total) return;
    const int h = t & (H - 1);
    const int e = t / H;
    const int s = (e < E) ? ei[e]     : (e - E);
    const int d = (e < E) ? ei[E + e] : (e - E);
    const float alpha = gat_leaky(a_s[s * H + h] + a_d[d * H + h]);
    atomicAdd(&denom[d * H + h], __expf(alpha - m[d * H + h]));
}

// ---------------------------------------------------------------------------
// Edge pass 3: out[dst] += softmax_weight * xp[src].  One wave per edge,
// each lane owns 4 of the 128 channels (float4 gather + 4 f32 atomics).
// ---------------------------------------------------------------------------
template <int H>
__global__ void gat_edge_agg_kernel(const int* __restrict__ ei, int E, int Nn,
                                    const float* __restrict__ a_s,
                                    const float* __restrict__ a_d,
                                    const float* __restrict__ m,
                                    const float* __restrict__ denom,
                                    const float* __restrict__ xp,
                                    float* __restrict__ outacc) {
    constexpr int C = GAT_HC / H;
    const long long t = (long long)blockIdx.x * blockDim.x + threadIdx.x;
    const int e = (int)(t >> 5);
    if (e >= E + Nn) return;
    const int lane = (int)(t & 31);
    const int c0 = lane * 4;              // 32 lanes x 4 = 128 channels
    const int h = c0 / C;                 // compile-time C -> shift
    const int s = (e < E) ? ei[e]     : (e - E);
    const int d = (e < E) ? ei[E + e] : (e - E);
    const float alpha = gat_leaky(a_s[s * H + h] + a_d[d * H + h]);
    const float w = __expf(alpha - m[d * H + h]) / denom[d * H + h];
    const float4 v = *(const float4*)(xp + (long long)s * GAT_HC + c0);
    float* o = outacc + (long long)d * GAT_HC + c0;
    atomicAdd(o + 0, w * v.x);
    atomicAdd(o + 1, w * v.y);
    atomicAdd(o + 2, w * v.z);
    atomicAdd(o + 3, w * v.w);
}

// ---------------------------------------------------------------------------
// Layer-1 finalize: h = elu(acc + b1), in place.
// ---------------------------------------------------------------------------
__global__ void gat_elu_bias_kernel(float* __restrict__ acc,
                                    const float* __restrict__ bias, int total) {
    int t = blockIdx.x * blockDim.x + threadIdx.x;
    if (t >= total) return;
    float v = acc[t] + bias[t & (GAT_HC - 1)];
    acc[t] = v > 0.f ? v : (__expf(v) - 1.f);
}

// ---------------------------------------------------------------------------
// Layer-2 finalize: out[n,c] = mean_h acc[n,h,c] + b2[c]   (H2=8, C2=16)
// ---------------------------------------------------------------------------
__global__ void gat_mean_bias_kernel(const float* __restrict__ acc,
                                     const float* __restrict__ bias,
                                     float* __restrict__ out, int Nn) {
    int t = blockIdx.x * blockDim.x + threadIdx.x;  // n*16 + c
    if (t >= Nn * 16) return;
    const int c = t & 15;
    const int n = t >> 4;
    const float* row = acc + (long long)n * GAT_HC;
    float sum = 0.f;
    #pragma unroll
    for (int h = 0; h < 8; ++h) sum += row[h * 16 + c];
    out[t] = sum * 0.125f + bias[c];
}

// ---------------------------------------------------------------------------
// Host-side launch sequence
// ---------------------------------------------------------------------------
extern "C" void kernel_launch(void* const* d_in, const int* in_sizes, int n_in,
                              void* d_out, int out_size, void* d_ws, size_t ws_size,
                              hipStream_t stream) {
    const float* x        = (const float*)d_in[0];
    const int*   ei       = (const int*)  d_in[1];   // [2,E] flat: src row then dst row
    // d_in[2] = edge_weight (ignored by reference)
    const float* W1       = (const float*)d_in[3];
    const float* att_src1 = (const float*)d_in[4];
    const float* att_dst1 = (const float*)d_in[5];
    const float* b1       = (const float*)d_in[6];
    const float* W2       = (const float*)d_in[7];
    const float* att_src2 = (const float*)d_in[8];
    const float* att_dst2 = (const float*)d_in[9];
    const float* b2       = (const float*)d_in[10];
    float*       out      = (float*)d_out;

    const int Nn = in_sizes[0] / 64;       // 100000
    const int E  = in_sizes[1] / 2;        // 1600000
    const int ET = E + Nn;                 // with self loops

    // Workspace carve-up (fp32), 256B aligned offsets
    char* ws = (char*)d_ws;
    size_t off = 0;
    auto carve = [&](size_t bytes) {
        float* p = (float*)(ws + off);
        off += (bytes + 255) & ~(size_t)255;
        return p;
    };
    float* B0    = carve((size_t)Nn * GAT_HC * 4);   // xp (both layers)
    float* B1    = carve((size_t)Nn * GAT_HC * 4);   // accumulator / h
    float* aS    = carve((size_t)Nn * 8 * 4);
    float* aD    = carve((size_t)Nn * 8 * 4);
    float* mBuf  = carve((size_t)Nn * 8 * 4);
    float* dBuf  = carve((size_t)Nn * 8 * 4);
    (void)ws_size;

    const int TB = 256;
    auto blocks = [&](long long work) { return (int)((work + TB - 1) / TB); };
    const float NEG_INF = -__builtin_huge_valf();

    const int tiles = ((Nn + 15) / 16) * (GAT_HC / 16);   // 16x16 output tiles
    const int gemmBlocks = (tiles + 7) / 8;               // 8 waves per 256-thr block

    // ----- Layer 1: H=4, C=32 -----
    {
        constexpr int H = 4;
        // xp1 = x @ W1   [Nn x 64] * [64 x 128]
        gat_wmma_gemm_f32<64><<<gemmBlocks, 256, 0, stream>>>(x, W1, B0, Nn);
        gat_att_kernel<H><<<blocks((long long)Nn * H), TB, 0, stream>>>(
            B0, att_src1, att_dst1, aS, aD, Nn);
        gat_fill_kernel<<<blocks((long long)Nn * H), TB, 0, stream>>>(mBuf, NEG_INF, Nn * H);
        gat_fill_kernel<<<blocks((long long)Nn * H), TB, 0, stream>>>(dBuf, 0.f, Nn * H);
        gat_fill_kernel<<<blocks((long long)Nn * GAT_HC), TB, 0, stream>>>(B1, 0.f, Nn * GAT_HC);
        gat_edge_max_kernel<H><<<blocks((long long)ET * H), TB, 0, stream>>>(
            ei, E, Nn, aS, aD, mBuf);
        gat_edge_sum_kernel<H><<<blocks((long long)ET * H), TB, 0, stream>>>(
            ei, E, Nn, aS, aD, mBuf, dBuf);
        gat_edge_agg_kernel<H><<<blocks((long long)ET * 32), TB, 0, stream>>>(
            ei, E, Nn, aS, aD, mBuf, dBuf, B0, B1);
        gat_elu_bias_kernel<<<blocks((long long)Nn * GAT_HC), TB, 0, stream>>>(
            B1, b1, Nn * GAT_HC);
    }

    // ----- Layer 2: H=8, C=16, mean over heads -----
    {
        constexpr int H = 8;
        // xp2 = h @ W2   [Nn x 128] * [128 x 128]
        gat_wmma_gemm_f32<128><<<gemmBlocks, 256, 0, stream>>>(B1, W2, B0, Nn);
        gat_att_kernel<H><<<blocks((long long)Nn * H), TB, 0, stream>>>(
            B0, att_src2, att_dst2, aS, aD, Nn);
        gat_fill_kernel<<<blocks((long long)Nn * H), TB, 0, stream>>>(mBuf, NEG_INF, Nn * H);
        gat_fill_kernel<<<blocks((long long)Nn * H), TB, 0, stream>>>(dBuf, 0.f, Nn * H);
        gat_fill_kernel<<<blocks((long long)Nn * GAT_HC), TB, 0, stream>>>(B1, 0.f, Nn * GAT_HC);
        gat_edge_max_kernel<H><<<blocks((long long)ET * H), TB, 0, stream>>>(
            ei, E, Nn, aS, aD, mBuf);
        gat_edge_sum_kernel<H><<<blocks((long long)ET * H), TB, 0, stream>>>(
            ei, E, Nn, aS, aD, mBuf, dBuf);
        gat_edge_agg_kernel<H><<<blocks((long long)ET * 32), TB, 0, stream>>>(
            ei, E, Nn, aS, aD, mBuf, dBuf, B0, B1);
        gat_mean_bias_kernel<<<blocks((long long)Nn * 16), TB, 0, stream>>>(
            B1, b2, out, Nn);
    }
    (void)out_size; (void)n_in;
}